// Matcher_83434034692462
// MI455X (gfx1250) — compile-verified
//
#include <hip/hip_runtime.h>
#include <hip/hip_bf16.h>
#include <math.h>

#define H 768
#define LN_EPS 1e-5f
#define L2_EPS 1e-12f
#define OT_INV_REG 20.0f   // 1 / 0.05
#define PAD 68             // LDS row stride (floats): 68*4=272B = 17*16B -> b128-aligned rows

typedef float v2f __attribute__((ext_vector_type(2)));
typedef float v8f __attribute__((ext_vector_type(8)));

// ---------------- CDNA5 async global->LDS copy (ASYNCcnt path) ----------------
// vdst = 32-bit LDS byte address (low 32 bits of the generic pointer = wave-relative
// LDS offset per ISA flat-aperture rules), vaddr = 64-bit global address, GV mode.
__device__ __forceinline__ void async_ld_b128(void* lds, const void* g) {
  asm volatile("global_load_async_to_lds_b128 %0, %1, off"
               :: "v"((unsigned int)(uintptr_t)lds),
                  "v"((unsigned long long)(uintptr_t)g)
               : "memory");
}
__device__ __forceinline__ void async_wait0() {
  asm volatile("s_wait_asynccnt 0" ::: "memory");
}

// ---------------- reductions ----------------
__device__ __forceinline__ float wave_sum(float v) {
  #pragma unroll
  for (int off = 16; off > 0; off >>= 1) v += __shfl_xor(v, off, 32);
  return v;
}
__device__ __forceinline__ float wave_max(float v) {
  #pragma unroll
  for (int off = 16; off > 0; off >>= 1) v = fmaxf(v, __shfl_xor(v, off, 32));
  return v;
}
// all 256 threads must participate
__device__ __forceinline__ float block_sum256(float v, float* red) {
  int lane = threadIdx.x & 31, w = threadIdx.x >> 5;
  v = wave_sum(v);
  if (lane == 0) red[w] = v;
  __syncthreads();
  float s = red[0] + red[1] + red[2] + red[3] + red[4] + red[5] + red[6] + red[7];
  __syncthreads();
  return s;
}

// ---------------- layernorm over rows of width 768 ----------------
__global__ void k_ln_rows(const float* __restrict__ x, const float* __restrict__ g,
                          const float* __restrict__ b, float* __restrict__ y, int rows) {
  __shared__ float red[8];
  int row = blockIdx.x;
  if (row >= rows) return;
  const float* xr = x + (size_t)row * H;
  int t = threadIdx.x;
  float v0 = xr[t], v1 = xr[t + 256], v2 = xr[t + 512];
  float mu = block_sum256(v0 + v1 + v2, red) * (1.f / H);
  float d0 = v0 - mu, d1 = v1 - mu, d2 = v2 - mu;
  float var = block_sum256(d0 * d0 + d1 * d1 + d2 * d2, red) * (1.f / H);
  float inv = 1.f / sqrtf(var + LN_EPS);
  float* yr = y + (size_t)row * H;
  yr[t]       = d0 * inv * g[t]       + b[t];
  yr[t + 256] = d1 * inv * g[t + 256] + b[t + 256];
  yr[t + 512] = d2 * inv * g[t + 512] + b[t + 512];
}

// ---------------- l2-normalize rows (in-place safe) ----------------
__global__ void k_l2n_rows(const float* __restrict__ x, float* __restrict__ y, int rows) {
  __shared__ float red[8];
  int row = blockIdx.x;
  if (row >= rows) return;
  const float* xr = x + (size_t)row * H;
  int t = threadIdx.x;
  float v0 = xr[t], v1 = xr[t + 256], v2 = xr[t + 512];
  float ss = block_sum256(v0 * v0 + v1 * v1 + v2 * v2, red);
  float inv = 1.f / fmaxf(sqrtf(ss), L2_EPS);
  float* yr = y + (size_t)row * H;
  yr[t] = v0 * inv; yr[t + 256] = v1 * inv; yr[t + 512] = v2 * inv;
}

// ---------------- out[M,768] = A[M,768] @ W[768,768] + bias ----------------
// block = 256 thr = 8 waves, tile 32(M) x 64(N), one 16x16 WMMA tile per wave.
// Tile staging via global_load_async_to_lds_b128 (all launched M are multiples of 32;
// row index clamped anyway so async reads stay in-bounds).
__global__ void k_gemm_bias(const float* __restrict__ A, const float* __restrict__ Wt,
                            const float* __restrict__ bias, float* __restrict__ out, int M) {
  __shared__ __align__(16) float As[32][PAD];
  __shared__ __align__(16) float Bs[64][PAD];
  const int tid = threadIdx.x, lane = tid & 31, w = tid >> 5;
  const int m0 = blockIdx.x * 32, n0 = blockIdx.y * 64;
  const int wm = (w & 1) << 4, wn = (w >> 1) << 4;
  v8f acc = {};
  const int am = wm + (lane & 15);
  const int bn = wn + (lane & 15);
  const int ksel = (lane >> 4) << 1;
  for (int kt = 0; kt < H; kt += 64) {
    // As: 32x64 floats = 512 x b128 chunks (2/thread)
    for (int i = tid; i < 512; i += 256) {
      int r = i >> 4, c4 = (i & 15) << 2;
      int gr = m0 + r; if (gr >= M) gr = M - 1;
      async_ld_b128(&As[r][c4], &A[(size_t)gr * H + kt + c4]);
    }
    // Bs: 64x64 floats = 1024 x b128 chunks (4/thread)
    for (int i = tid; i < 1024; i += 256) {
      int r = i >> 4, c4 = (i & 15) << 2;
      async_ld_b128(&Bs[r][c4], &Wt[(size_t)(kt + r) * H + n0 + c4]);
    }
    async_wait0();
    __syncthreads();
    #pragma unroll
    for (int kk = 0; kk < 64; kk += 4) {
      int k0 = kk + ksel;
      v2f a, b;
      a.x = As[am][k0]; a.y = As[am][k0 + 1];
      b.x = Bs[k0][bn]; b.y = Bs[k0 + 1][bn];
      acc = __builtin_amdgcn_wmma_f32_16x16x4_f32(false, a, false, b, (short)0, acc, false, false);
    }
    __syncthreads();
  }
  const int col = n0 + wn + (lane & 15);
  const float bv = bias[col];
  #pragma unroll
  for (int r = 0; r < 8; ++r) {
    int row = m0 + wm + ((lane >> 4) << 3) + r;
    if (row < M) out[(size_t)row * H + col] = acc[r] + bv;
  }
}

// ---------------- batched C[p,i,j] = dot(Arow(p,i), Brow(p,j)) over 768 ----------------
// bCount>0: cross pairs (a=p/bCount, b=p%bCount); bCount==0: aligned (a=b=p).
// Out-of-range rows clamped (finite garbage only feeds masked output rows/cols).
__global__ void k_batched_abt(const float* __restrict__ Abase, const float* __restrict__ Bbase,
                              float* __restrict__ C, int Sa, int Sb, int bCount) {
  __shared__ __align__(16) float As[64][PAD];
  __shared__ __align__(16) float Bs[64][PAD];
  int p = blockIdx.x;
  int ai = (bCount > 0) ? (p / bCount) : p;
  int bi = (bCount > 0) ? (p % bCount) : p;
  const float* A = Abase + (size_t)ai * Sa * H;
  const float* B = Bbase + (size_t)bi * Sb * H;
  int tid = threadIdx.x, lane = tid & 31, w = tid >> 5;
  v8f acc[2] = {};
  const int ksel = (lane >> 4) << 1;
  for (int kt = 0; kt < H; kt += 64) {
    for (int i = tid; i < 1024; i += 256) {
      int r = i >> 4, c4 = (i & 15) << 2;
      int ra = (r < Sa) ? r : Sa - 1;
      int rb = (r < Sb) ? r : Sb - 1;
      async_ld_b128(&As[r][c4], &A[(size_t)ra * H + kt + c4]);
      async_ld_b128(&Bs[r][c4], &B[(size_t)rb * H + kt + c4]);
    }
    async_wait0();
    __syncthreads();
    #pragma unroll
    for (int kk = 0; kk < 64; kk += 4) {
      int k0 = kk + ksel;
      #pragma unroll
      for (int tt = 0; tt < 2; ++tt) {
        int t = w + (tt << 3);
        int tm = (t & 3) << 4, tn = (t >> 2) << 4;
        v2f a, b;
        a.x = As[tm + (lane & 15)][k0]; a.y = As[tm + (lane & 15)][k0 + 1];
        b.x = Bs[tn + (lane & 15)][k0]; b.y = Bs[tn + (lane & 15)][k0 + 1];
        acc[tt] = __builtin_amdgcn_wmma_f32_16x16x4_f32(false, a, false, b, (short)0, acc[tt], false, false);
      }
    }
    __syncthreads();
  }
  #pragma unroll
  for (int tt = 0; tt < 2; ++tt) {
    int t = w + (tt << 3);
    int tm = (t & 3) << 4, tn = (t >> 2) << 4;
    int j = tn + (lane & 15);
    #pragma unroll
    for (int r = 0; r < 8; ++r) {
      int i = tm + ((lane >> 4) << 3) + r;
      if (i < Sa && j < Sb) C[((size_t)p * Sa + i) * Sb + j] = acc[tt][r];
    }
  }
}

// ---------------- sinkhorn (5 iters) + row l2n, in place on C[p,Sa,Sb] ----------------
__global__ void k_sinkhorn_l2n(float* __restrict__ C, int Sa, int Sb) {
  __shared__ float sm[49 * 50 + 64];
  float* Q = sm;
  float* rs = sm + 49 * 50;
  int p = blockIdx.x, tid = threadIdx.x;
  float* Cg = C + (size_t)p * Sa * Sb;
  int total = Sa * Sb, W = Sb + 1;
  for (int i = tid; i < total; i += 256) {
    int r = i / Sb, c = i - r * Sb;
    Q[r * W + c] = expf(Cg[i] * OT_INV_REG);
  }
  __syncthreads();
  for (int it = 0; it < 5; ++it) {
    if (tid < Sa) { float s = 0; for (int j = 0; j < Sb; ++j) s += Q[tid * W + j]; rs[tid] = s; }
    __syncthreads();
    for (int i = tid; i < total; i += 256) { int r = i / Sb, c = i - r * Sb; Q[r * W + c] /= rs[r]; }
    __syncthreads();
    if (tid < Sb) { float s = 0; for (int r = 0; r < Sa; ++r) s += Q[r * W + tid]; rs[tid] = s; }
    __syncthreads();
    for (int i = tid; i < total; i += 256) { int r = i / Sb, c = i - r * Sb; Q[r * W + c] /= rs[c]; }
    __syncthreads();
  }
  if (tid < Sa) {
    float s = 0;
    for (int j = 0; j < Sb; ++j) { float q = Q[tid * W + j]; s += q * q; }
    rs[tid] = 1.f / fmaxf(sqrtf(s), L2_EPS);
  }
  __syncthreads();
  for (int i = tid; i < total; i += 256) { int r = i / Sb, c = i - r * Sb; Cg[i] = Q[r * W + c] * rs[r]; }
}

// ---------------- fused uni-matcher tail: per (a,b) pair ----------------
// ctx = P@v + q (LDS-resident, WMMA), row-LN(cross_ln), softpool, dot(ecls) -> g2l[b*N+a].
// Ps K-padding stays explicitly zero so clamped Vs rows multiply by 0.
__global__ void k_uni_fused(const float* __restrict__ P, const float* __restrict__ vmat,
                            const float* __restrict__ q, const float* __restrict__ ecls,
                            const float* __restrict__ lng, const float* __restrict__ lnb,
                            const float* __restrict__ pw, const float* __restrict__ pb1,
                            const float* __restrict__ plg, const float* __restrict__ plb,
                            float* __restrict__ g2l, int Sa, int Sb, int Bcnt, int Ncnt) {
  extern __shared__ float sm[];
  float* ctx = sm;                      // Sa*768
  float* Ps  = ctx + (size_t)Sa * H;    // 64*PAD zero-padded
  float* Vs  = Ps + 64 * PAD;           // 64*PAD (16B-aligned: all offsets multiple of 4 floats)
  float* sA  = Vs + 64 * PAD;           // 64
  float* wA  = sA + 64;                 // 64
  float* red = wA + 64;                 // 8
  float* pooled = red + 8;              // 768
  int p = blockIdx.x;
  int a = p / Bcnt, b = p % Bcnt;
  const float* Pg = P + (size_t)p * Sa * Sb;
  const float* vb = vmat + (size_t)b * Sb * H;
  const float* qa = q + (size_t)a * Sa * H;
  int tid = threadIdx.x, lane = tid & 31, w = tid >> 5;
  const int ksel = (lane >> 4) << 1;

  for (int i = tid; i < 64 * PAD; i += 256) Ps[i] = 0.f;
  __syncthreads();
  for (int i = tid; i < Sa * Sb; i += 256) { int r = i / Sb, c = i - r * Sb; Ps[r * PAD + c] = Pg[i]; }

  for (int dt = 0; dt < H; dt += 64) {
    for (int i = tid; i < 1024; i += 256) {
      int r = i >> 4, c4 = (i & 15) << 2;
      int rc = (r < Sb) ? r : Sb - 1;
      async_ld_b128(&Vs[r * PAD + c4], &vb[(size_t)rc * H + dt + c4]);
    }
    async_wait0();
    __syncthreads();
    v8f acc[2] = {};
    #pragma unroll
    for (int kk = 0; kk < 64; kk += 4) {
      int k0 = kk + ksel;
      #pragma unroll
      for (int tt = 0; tt < 2; ++tt) {
        int t = w + (tt << 3);
        int tm = (t & 3) << 4, tn = (t >> 2) << 4;
        v2f av, bv;
        av.x = Ps[(tm + (lane & 15)) * PAD + k0]; av.y = Ps[(tm + (lane & 15)) * PAD + k0 + 1];
        bv.x = Vs[k0 * PAD + tn + (lane & 15)];   bv.y = Vs[(k0 + 1) * PAD + tn + (lane & 15)];
        acc[tt] = __builtin_amdgcn_wmma_f32_16x16x4_f32(false, av, false, bv, (short)0, acc[tt], false, false);
      }
    }
    __syncthreads();   // all waves done reading Vs before next async overwrite
    #pragma unroll
    for (int tt = 0; tt < 2; ++tt) {
      int t = w + (tt << 3);
      int tm = (t & 3) << 4, tn = (t >> 2) << 4;
      int n = tn + (lane & 15);
      #pragma unroll
      for (int r = 0; r < 8; ++r) {
        int m = tm + ((lane >> 4) << 3) + r;
        if (m < Sa) ctx[(size_t)m * H + dt + n] = acc[tt][r] + qa[(size_t)m * H + dt + n];
      }
    }
  }
  __syncthreads();

  // per-row LayerNorm (cross_ln), one wave per row
  for (int i = w; i < Sa; i += 8) {
    float s = 0;
    for (int t = lane; t < H; t += 32) s += ctx[(size_t)i * H + t];
    float mu = wave_sum(s) * (1.f / H);
    float sq = 0;
    for (int t = lane; t < H; t += 32) { float d = ctx[(size_t)i * H + t] - mu; sq += d * d; }
    float inv = 1.f / sqrtf(wave_sum(sq) * (1.f / H) + LN_EPS);
    for (int t = lane; t < H; t += 32)
      ctx[(size_t)i * H + t] = (ctx[(size_t)i * H + t] - mu) * inv * lng[t] + lnb[t];
  }
  __syncthreads();
  // softpool scores
  for (int i = w; i < Sa; i += 8) {
    float s = 0;
    for (int t = lane; t < H; t += 32) s += ctx[(size_t)i * H + t] * pw[t];
    s = wave_sum(s);
    if (lane == 0) sA[i] = s + pb1[0];
  }
  __syncthreads();
  if (w == 0) {
    float mx = -1e30f;
    for (int i = lane; i < Sa; i += 32) mx = fmaxf(mx, sA[i]);
    mx = wave_max(mx);
    float sum = 0;
    for (int i = lane; i < Sa; i += 32) { float e = expf(sA[i] - mx); wA[i] = e; sum += e; }
    sum = wave_sum(sum);
    for (int i = lane; i < Sa; i += 32) wA[i] /= sum;
  }
  __syncthreads();
  for (int d = tid; d < H; d += 256) {
    float s = 0;
    for (int i = 0; i < Sa; ++i) s += ctx[(size_t)i * H + d] * wA[i];
    pooled[d] = s;
  }
  __syncthreads();
  float v0 = pooled[tid], v1 = pooled[tid + 256], v2 = pooled[tid + 512];
  float mu = block_sum256(v0 + v1 + v2, red) * (1.f / H);
  float d0 = v0 - mu, d1 = v1 - mu, d2 = v2 - mu;
  float var = block_sum256(d0 * d0 + d1 * d1 + d2 * d2, red) * (1.f / H);
  float inv = 1.f / sqrtf(var + LN_EPS);
  float y0 = d0 * inv * plg[tid] + plb[tid];
  float y1 = d1 * inv * plg[tid + 256] + plb[tid + 256];
  float y2 = d2 * inv * plg[tid + 512] + plb[tid + 512];
  const float* ea = ecls + (size_t)a * H;
  float dot = y0 * ea[tid] + y1 * ea[tid + 256] + y2 * ea[tid + 512];
  dot = block_sum256(dot, red);
  if (tid == 0) g2l[b * Ncnt + a] = dot;
}

// ---------------- ot_transport tail: out[row] = LN(P[row,:] @ o_tr + t[row]) ----------------
__global__ void k_ot_out(const float* __restrict__ P, const float* __restrict__ otr,
                         const float* __restrict__ t, const float* __restrict__ lng,
                         const float* __restrict__ lnb, float* __restrict__ out,
                         int St, int So) {
  __shared__ float Prow[64];
  __shared__ float red[8];
  int row = blockIdx.x;
  int bidx = row / St;
  const float* Pg = P + (size_t)row * So;
  for (int j = threadIdx.x; j < So; j += 256) Prow[j] = Pg[j];
  __syncthreads();
  const float* ob = otr + (size_t)bidx * So * H;
  const float* tr = t + (size_t)row * H;
  int tid = threadIdx.x;
  float acc[3];
  #pragma unroll
  for (int e = 0; e < 3; ++e) {
    int d = tid + e * 256;
    float s = tr[d];
    for (int j = 0; j < So; ++j) s += Prow[j] * ob[(size_t)j * H + d];
    acc[e] = s;
  }
  float mu = block_sum256(acc[0] + acc[1] + acc[2], red) * (1.f / H);
  float d0 = acc[0] - mu, d1 = acc[1] - mu, d2 = acc[2] - mu;
  float var = block_sum256(d0 * d0 + d1 * d1 + d2 * d2, red) * (1.f / H);
  float inv = 1.f / sqrtf(var + LN_EPS);
  float* o = out + (size_t)row * H;
  o[tid]       = d0 * inv * lng[tid]       + lnb[tid];
  o[tid + 256] = d1 * inv * lng[tid + 256] + lnb[tid + 256];
  o[tid + 512] = d2 * inv * lng[tid + 512] + lnb[tid + 512];
}

// ---------------- softpool over concat([X1, X2], seq axis), then LN ----------------
__global__ void k_softpool2(const float* __restrict__ X1, int S1,
                            const float* __restrict__ X2, int S2,
                            const float* __restrict__ pw, const float* __restrict__ pb1,
                            const float* __restrict__ lng, const float* __restrict__ lnb,
                            float* __restrict__ out) {
  __shared__ float sA[128], wA[128], red[8];
  int p = blockIdx.x, tid = threadIdx.x, lane = tid & 31, w = tid >> 5;
  int Sc = S1 + S2;
  const float* x1 = X1 + (size_t)p * S1 * H;
  const float* x2 = X2 + (size_t)p * S2 * H;
  for (int i = w; i < Sc; i += 8) {
    const float* xr = (i < S1) ? (x1 + (size_t)i * H) : (x2 + (size_t)(i - S1) * H);
    float s = 0;
    for (int t = lane; t < H; t += 32) s += xr[t] * pw[t];
    s = wave_sum(s);
    if (lane == 0) sA[i] = s + pb1[0];
  }
  __syncthreads();
  if (w == 0) {
    float mx = -1e30f;
    for (int i = lane; i < Sc; i += 32) mx = fmaxf(mx, sA[i]);
    mx = wave_max(mx);
    float sum = 0;
    for (int i = lane; i < Sc; i += 32) { float e = expf(sA[i] - mx); wA[i] = e; sum += e; }
    sum = wave_sum(sum);
    for (int i = lane; i < Sc; i += 32) wA[i] /= sum;
  }
  __syncthreads();
  float a0 = 0, a1 = 0, a2 = 0;
  for (int i = 0; i < Sc; ++i) {
    const float* xr = (i < S1) ? (x1 + (size_t)i * H) : (x2 + (size_t)(i - S1) * H);
    float wi = wA[i];
    a0 += xr[tid] * wi; a1 += xr[tid + 256] * wi; a2 += xr[tid + 512] * wi;
  }
  float mu = block_sum256(a0 + a1 + a2, red) * (1.f / H);
  float d0 = a0 - mu, d1 = a1 - mu, d2 = a2 - mu;
  float var = block_sum256(d0 * d0 + d1 * d1 + d2 * d2, red) * (1.f / H);
  float inv = 1.f / sqrtf(var + LN_EPS);
  float* o = out + (size_t)p * H;
  o[tid]       = d0 * inv * lng[tid]       + lnb[tid];
  o[tid + 256] = d1 * inv * lng[tid + 256] + lnb[tid + 256];
  o[tid + 512] = d2 * inv * lng[tid + 512] + lnb[tid + 512];
}

// ---------------- final combine: d_out = [score | txt | img | mm], each [32,64] ----------------
__global__ void k_final(const float* __restrict__ g2l_t, const float* __restrict__ g2g_t,
                        const float* __restrict__ g2l_i, const float* __restrict__ g2g_i,
                        const float* __restrict__ mmp0, const float* __restrict__ mmp1,
                        float* __restrict__ out) {
  int idx = blockIdx.x * 256 + threadIdx.x;
  if (idx >= 2048) return;
  float t  = 0.5f * (g2l_t[idx] + g2g_t[idx]);
  float im = 0.5f * (g2l_i[idx] + g2g_i[idx]);
  float mm = 0.5f * (mmp0[idx] + mmp1[idx]);
  out[idx]        = (t + im + mm) * (1.f / 3.f);
  out[2048 + idx] = t;
  out[4096 + idx] = im;
  out[6144 + idx] = mm;
}

// =====================================================================
// Host orchestration. Params = jax pytree leaves (dict keys alphabetical)
// appended after the 8 input tensors; index map in comments below.
// =====================================================================
extern "C" void kernel_launch(void* const* d_in, const int* in_sizes, int n_in,
                              void* d_out, int out_size, void* d_ws, size_t ws_size,
                              hipStream_t stream) {
  (void)in_sizes; (void)n_in; (void)out_size; (void)ws_size;
  #define IN(i) ((const float*)d_in[i])
  const int N = 64, B = 32, ST = 48, SI = 49;
  const int RTe = N * ST, RTm = B * ST, RIe = N * SI, RIm = B * SI;
  const dim3 blk(256);

  float* Wp = (float*)d_ws;
  size_t off = 0;
  auto alloc = [&](size_t n) { float* p = Wp + off; off += n; return p; };

  // persistent
  float* e_txt = alloc((size_t)RTe * H);
  float* m_txt = alloc((size_t)RTm * H);
  float* e_img = alloc((size_t)RIe * H);
  float* m_img = alloc((size_t)RIm * H);
  float* e_tcls = alloc((size_t)N * H);
  float* m_tcls = alloc((size_t)B * H);
  float* e_icls = alloc((size_t)N * H);
  float* m_icls = alloc((size_t)B * H);
  float* g2l_t = alloc(2048); float* g2g_t = alloc(2048);
  float* g2l_i = alloc(2048); float* g2g_i = alloc(2048);
  float* mmp0  = alloc(2048); float* mmp1  = alloc(2048);
  const size_t mark = off;

  // input layernorms: ln_text_tok g=29 b=28; ln_img_tok g=25 b=24;
  // ln_text_cls g=27 b=26; ln_img_cls g=23 b=22
  k_ln_rows<<<RTe, blk, 0, stream>>>(IN(0), IN(29), IN(28), e_txt, RTe);
  k_ln_rows<<<RTm, blk, 0, stream>>>(IN(1), IN(29), IN(28), m_txt, RTm);
  k_ln_rows<<<RIe, blk, 0, stream>>>(IN(2), IN(25), IN(24), e_img, RIe);
  k_ln_rows<<<RIm, blk, 0, stream>>>(IN(3), IN(25), IN(24), m_img, RIm);
  k_ln_rows<<<N,   blk, 0, stream>>>(IN(4), IN(27), IN(26), e_tcls, N);
  k_ln_rows<<<B,   blk, 0, stream>>>(IN(5), IN(27), IN(26), m_tcls, B);
  k_ln_rows<<<N,   blk, 0, stream>>>(IN(6), IN(23), IN(22), e_icls, N);
  k_ln_rows<<<B,   blk, 0, stream>>>(IN(7), IN(23), IN(22), m_icls, B);

  // generic uni-matcher phase. param idx: {clnB,clnG,fcClsB,fcClsW,fcKb,fcKw,fcQb,fcQw,fcVb,fcVw,pfB,pfW,plnB,plnG}
  auto uni = [&](const float* ecls_ln, const float* etok, const float* mtok,
                 int S, int Re, int Rm, const int pi[14], float* g2l) {
    off = mark;
    float* eclsp = alloc((size_t)N * H);
    float* q  = alloc((size_t)Re * H);
    float* k  = alloc((size_t)Rm * H);
    float* v  = alloc((size_t)Rm * H);
    float* qn = alloc((size_t)Re * H);
    float* kn = alloc((size_t)Rm * H);
    float* C  = alloc((size_t)N * B * S * S);
    k_gemm_bias<<<dim3((N + 31) / 32, 12), blk, 0, stream>>>(ecls_ln, IN(pi[3]), IN(pi[2]), eclsp, N);
    k_gemm_bias<<<dim3(Re / 32, 12), blk, 0, stream>>>(etok, IN(pi[7]), IN(pi[6]), q, Re);
    k_gemm_bias<<<dim3(Rm / 32, 12), blk, 0, stream>>>(mtok, IN(pi[5]), IN(pi[4]), k, Rm);
    k_gemm_bias<<<dim3(Rm / 32, 12), blk, 0, stream>>>(mtok, IN(pi[9]), IN(pi[8]), v, Rm);
    k_l2n_rows<<<Re, blk, 0, stream>>>(q, qn, Re);
    k_l2n_rows<<<Rm, blk, 0, stream>>>(k, kn, Rm);
    k_batched_abt<<<N * B, blk, 0, stream>>>(qn, kn, C, S, S, B);
    k_sinkhorn_l2n<<<N * B, blk, 0, stream>>>(C, S, S);
    size_t shm = ((size_t)S * H + 2 * 64 * PAD + 64 + 64 + 8 + H) * sizeof(float);
    k_uni_fused<<<N * B, blk, shm, stream>>>(C, v, q, eclsp, IN(pi[1]), IN(pi[0]),
                                             IN(pi[11]), IN(pi[10]), IN(pi[13]), IN(pi[12]),
                                             g2l, S, S, B, N);
  };
  const int piT[14] = {66, 67, 68, 69, 70, 71, 72, 73, 74, 75, 76, 77, 78, 79};
  const int piI[14] = {8, 9, 10, 11, 12, 13, 14, 15, 16, 17, 18, 19, 20, 21};
  uni(e_tcls, e_txt, m_txt, ST, RTe, RTm, piT, g2l_t);
  k_batched_abt<<<1, blk, 0, stream>>>(m_tcls, e_tcls, g2g_t, B, N, 0);
  uni(e_icls, e_img, m_img, SI, RIe, RIm, piI, g2l_i);
  k_batched_abt<<<1, blk, 0, stream>>>(m_icls, e_icls, g2g_i, B, N, 0);

  // ---------------- mm matcher ----------------
  off = mark;
  float* e_ft = alloc((size_t)RTe * H);
  float* m_ft = alloc((size_t)RTm * H);
  float* e_fi = alloc((size_t)RIe * H);
  float* m_fi = alloc((size_t)RIm * H);
  float* e_img_tr = alloc((size_t)RTe * H);  // [64,48,768]
  float* e_txt_tr = alloc((size_t)RIe * H);  // [64,49,768]
  float* m_img_tr = alloc((size_t)RTm * H);
  float* m_txt_tr = alloc((size_t)RIm * H);
  float* e0 = alloc((size_t)N * H); float* e1 = alloc((size_t)N * H);
  float* m0 = alloc((size_t)B * H); float* m1 = alloc((size_t)B * H);
  const size_t otmark = off;

  // ot param idx: {toB,toW, ttB,ttW, ftB,ftW, foB,foW, fotB,fotW, lnB,lnG}
  auto ot = [&](const int pi[12], const float* origin, int So,
                const float* target, int St, int batch, float* outp) {
    off = otmark;
    int Ro = batch * So, Rt = batch * St;
    float* o   = alloc((size_t)Ro * H);
    float* t1  = alloc((size_t)Rt * H);
    float* tt  = alloc((size_t)Rt * H);
    float* tn  = alloc((size_t)Rt * H);
    float* ofc = alloc((size_t)Ro * H);
    float* otr = alloc((size_t)Ro * H);
    float* Cmm = alloc((size_t)batch * St * So);
    k_gemm_bias<<<dim3(Ro / 32, 12), blk, 0, stream>>>(origin, IN(pi[1]), IN(pi[0]), o, Ro);
    k_gemm_bias<<<dim3(Rt / 32, 12), blk, 0, stream>>>(target, IN(pi[3]), IN(pi[2]), t1, Rt);
    k_gemm_bias<<<dim3(Rt / 32, 12), blk, 0, stream>>>(t1, IN(pi[5]), IN(pi[4]), tt, Rt);
    k_gemm_bias<<<dim3(Ro / 32, 12), blk, 0, stream>>>(o, IN(pi[7]), IN(pi[6]), ofc, Ro);
    k_gemm_bias<<<dim3(Ro / 32, 12), blk, 0, stream>>>(o, IN(pi[9]), IN(pi[8]), otr, Ro);
    k_l2n_rows<<<Rt, blk, 0, stream>>>(tt, tn, Rt);
    k_l2n_rows<<<Ro, blk, 0, stream>>>(ofc, ofc, Ro);
    k_batched_abt<<<batch, blk, 0, stream>>>(tn, ofc, Cmm, St, So, 0);
    k_sinkhorn_l2n<<<batch, blk, 0, stream>>>(Cmm, St, So);
    k_ot_out<<<Rt, blk, 0, stream>>>(Cmm, otr, tt, IN(pi[11]), IN(pi[10]), outp, St, So);
  };
  // i2t: trans_ori 42/43, trans_tar 44/45, fc_target 38/39, fc_origin 34/35, fc_or_tr 36/37, ln 40/41
  const int piI2T[12] = {42, 43, 44, 45, 38, 39, 34, 35, 36, 37, 40, 41};
  const int piT2I[12] = {62, 63, 64, 65, 58, 59, 54, 55, 56, 57, 60, 61};
  ot(piI2T, e_img, SI, e_txt, ST, N, e_img_tr);
  ot(piT2I, e_txt, ST, e_img, SI, N, e_txt_tr);
  ot(piI2T, m_img, SI, m_txt, ST, B, m_img_tr);
  ot(piT2I, m_txt, ST, m_img, SI, B, m_txt_tr);

  // fc_text 32/33, fc_image 30/31
  k_gemm_bias<<<dim3(RTe / 32, 12), blk, 0, stream>>>(e_txt, IN(33), IN(32), e_ft, RTe);
  k_gemm_bias<<<dim3(RTm / 32, 12), blk, 0, stream>>>(m_txt, IN(33), IN(32), m_ft, RTm);
  k_gemm_bias<<<dim3(RIe / 32, 12), blk, 0, stream>>>(e_img, IN(31), IN(30), e_fi, RIe);
  k_gemm_bias<<<dim3(RIm / 32, 12), blk, 0, stream>>>(m_img, IN(31), IN(30), m_fi, RIm);

  // pool0: fcB 46 fcW 47 lnB 48 lnG 49; pool1: 50/51/52/53
  k_softpool2<<<N, blk, 0, stream>>>(e_ft, ST, e_img_tr, ST, IN(47), IN(46), IN(49), IN(48), e0);
  k_softpool2<<<N, blk, 0, stream>>>(e_txt_tr, SI, e_fi, SI, IN(51), IN(50), IN(53), IN(52), e1);
  k_softpool2<<<B, blk, 0, stream>>>(m_ft, ST, m_img_tr, ST, IN(47), IN(46), IN(49), IN(48), m0);
  k_softpool2<<<B, blk, 0, stream>>>(m_txt_tr, SI, m_fi, SI, IN(51), IN(50), IN(53), IN(52), m1);

  k_batched_abt<<<1, blk, 0, stream>>>(m0, e0, mmp0, B, N, 0);
  k_batched_abt<<<1, blk, 0, stream>>>(m1, e1, mmp1, B, N, 0);

  k_final<<<8, blk, 0, stream>>>(g2l_t, g2g_t, g2l_i, g2g_i, mmp0, mmp1, (float*)d_out);
  #undef IN
}